// Model_42408507081126
// MI455X (gfx1250) — compile-verified
//
#include <hip/hip_runtime.h>
#include <hip/hip_bf16.h>

typedef float v2f __attribute__((ext_vector_type(2)));
typedef float v8f __attribute__((ext_vector_type(8)));

#define GN   100000
#define GE   400000
#define GB   512
#define GD   300
#define GD4  75        // GD / 4
#define HID  600
#define W1PAD 608      // 600 padded to 38 tiles (19 tile-pairs)
#define W2PAD 320      // 300 padded to 20 tiles (10 tile-pairs)
#define INV_TEMP 25.0f
#define BNEPS 1e-5f

#define TILE_ROWS 32
#define A_STRIDE  308   // 300 padded (stride%64 = 52 -> conflict-free frag reads)
#define H_STRIDE  604   // 600 padded (stride%64 = 28 -> conflict-free frag reads)

// ---------------------------------------------------------------- utilities

__global__ void zero_kernel(float* p, int n) {
    int i = blockIdx.x * blockDim.x + threadIdx.x;
    if (i < n) p[i] = 0.0f;
}

// Wt[c*K + k] = (c < Ncol) ? W[k*Ncol + c] : 0   (transpose + zero-pad columns)
__global__ void pack_kernel(const float* __restrict__ W, float* __restrict__ Wt,
                            int K, int Ncol, int NcolPad) {
    int tid = blockIdx.x * blockDim.x + threadIdx.x;
    if (tid >= NcolPad * K) return;
    int c = tid / K, k = tid % K;
    Wt[tid] = (c < Ncol) ? W[(size_t)k * Ncol + c] : 0.0f;
}

// ---------------------------------------------------------------- embeddings

__global__ void embed_kernel(const int* __restrict__ x,
                             const float* __restrict__ xe1,
                             const float* __restrict__ xe2,
                             float* __restrict__ h) {
    int tid = blockIdx.x * blockDim.x + threadIdx.x;
    if (tid >= GN * GD4) return;
    int n = tid / GD4, j = tid % GD4;
    int a = x[2 * n], b = x[2 * n + 1];
    float4 v1 = ((const float4*)xe1)[(size_t)a * GD4 + j];
    float4 v2 = ((const float4*)xe2)[(size_t)b * GD4 + j];
    float4 o;
    o.x = v1.x + v2.x; o.y = v1.y + v2.y; o.z = v1.z + v2.z; o.w = v1.w + v2.w;
    ((float4*)h)[(size_t)n * GD4 + j] = o;
}

// agg[n] = h[n] + ee1[4] + ee2[0]   (self-loop contribution, also zero-init)
__global__ void init_agg_kernel(const float* __restrict__ h,
                                const float* __restrict__ ee1,
                                const float* __restrict__ ee2,
                                float* __restrict__ agg) {
    int tid = blockIdx.x * blockDim.x + threadIdx.x;
    if (tid >= GN * GD4) return;
    int n = tid / GD4, j = tid % GD4;
    float4 hv = ((const float4*)h)[(size_t)n * GD4 + j];
    float4 e1 = ((const float4*)ee1)[4 * GD4 + j];
    float4 e2 = ((const float4*)ee2)[j];
    float4 o;
    o.x = hv.x + e1.x + e2.x; o.y = hv.y + e1.y + e2.y;
    o.z = hv.z + e1.z + e2.z; o.w = hv.w + e1.w + e2.w;
    ((float4*)agg)[(size_t)n * GD4 + j] = o;
}

// scatter-add of messages over real edges
__global__ void edge_scatter_kernel(const int* __restrict__ ei,
                                    const int* __restrict__ ea,
                                    const float* __restrict__ h,
                                    const float* __restrict__ ee1,
                                    const float* __restrict__ ee2,
                                    float* __restrict__ agg) {
    int tid = blockIdx.x * blockDim.x + threadIdx.x;
    if (tid >= GE * GD4) return;
    int e = tid / GD4, j = tid % GD4;
    int s  = ei[e], d = ei[GE + e];
    int a0 = ea[2 * e], a1 = ea[2 * e + 1];
    float4 hv = ((const float4*)h)[(size_t)s * GD4 + j];
    float4 e1 = ((const float4*)ee1)[(size_t)a0 * GD4 + j];
    float4 e2 = ((const float4*)ee2)[(size_t)a1 * GD4 + j];
    float* dst = agg + (size_t)d * GD + 4 * j;
    unsafeAtomicAdd(dst + 0, hv.x + e1.x + e2.x);
    unsafeAtomicAdd(dst + 1, hv.y + e1.y + e2.y);
    unsafeAtomicAdd(dst + 2, hv.z + e1.z + e2.z);
    unsafeAtomicAdd(dst + 3, hv.w + e1.w + e2.w);
}

// ---------------------------------------------------------------- fused GIN MLP
// hout = relu(agg @ W1 + b1) @ W2 + b2, fused through LDS, fp32 WMMA 16x16x4.
// Weights pre-packed transposed + column-padded: W1t[608][300], W2t[320][600].

__global__ __launch_bounds__(256)
void gin_mlp_kernel(const float* __restrict__ agg,
                    const float* __restrict__ W1t, const float* __restrict__ b1,
                    const float* __restrict__ W2t, const float* __restrict__ b2,
                    float* __restrict__ hout) {
    __shared__ float aTile[TILE_ROWS * A_STRIDE];   // 32 x 300 agg block
    __shared__ float hTile[TILE_ROWS * H_STRIDE];   // 32 x 600 hidden block

    const int row0  = blockIdx.x * TILE_ROWS;
    const int tid   = threadIdx.x;
    const int lane  = tid & 31;
    const int wave  = tid >> 5;        // 0..7
    const int lhalf = lane >> 4;       // 0 / 1
    const int l16   = lane & 15;

    // cooperatively stage A block (coalesced; N % 32 == 0, no guard needed)
    for (int idx = tid; idx < TILE_ROWS * GD; idx += 256) {
        int r = idx / GD, c = idx % GD;
        aTile[r * A_STRIDE + c] = agg[(size_t)(row0 + r) * GD + c];
    }
    __syncthreads();

    // GEMM1: hidden = relu(A[32x300] @ W1[300x600] + b1)
    // units: 2 row-tiles x 19 column-tile-pairs, 16x32 strip per unit
    for (int unit = wave; unit < 2 * 19; unit += 8) {
        int rt = unit / 19, cp = unit % 19;
        int col0 = cp * 32 + l16;
        int col1 = col0 + 16;
        const float* aBase = &aTile[(rt * 16 + l16) * A_STRIDE];
        const float* bB0 = W1t + (size_t)col0 * GD;
        const float* bB1 = W1t + (size_t)col1 * GD;
        v8f acc0 = {}, acc1 = {};
        for (int k = 0; k < GD; k += 4) {
            int ka = k + 2 * lhalf;
            v2f a  = *(const v2f*)(aBase + ka);   // ds_load_b64
            v2f bv0 = *(const v2f*)(bB0 + ka);    // global_load_b64
            v2f bv1 = *(const v2f*)(bB1 + ka);    // global_load_b64
            acc0 = __builtin_amdgcn_wmma_f32_16x16x4_f32(
                       false, a, false, bv0, (short)0, acc0, false, false);
            acc1 = __builtin_amdgcn_wmma_f32_16x16x4_f32(
                       false, a, false, bv1, (short)0, acc1, false, false);
        }
        {   // col0 < 592+8 = always < 600: unconditional
            float bias = b1[col0];
            for (int j = 0; j < 8; ++j) {
                int rloc = rt * 16 + j + 8 * lhalf;   // C/D: VGPR j -> rows j / j+8
                float v = acc0[j] + bias;
                hTile[rloc * H_STRIDE + col0] = v > 0.0f ? v : 0.0f;
            }
        }
        if (col1 < HID) {
            float bias = b1[col1];
            for (int j = 0; j < 8; ++j) {
                int rloc = rt * 16 + j + 8 * lhalf;
                float v = acc1[j] + bias;
                hTile[rloc * H_STRIDE + col1] = v > 0.0f ? v : 0.0f;
            }
        }
    }
    __syncthreads();

    // GEMM2: out = hidden[32x600] @ W2[600x300] + b2
    // units: 2 row-tiles x 10 column-tile-pairs
    for (int unit = wave; unit < 2 * 10; unit += 8) {
        int rt = unit / 10, cp = unit % 10;
        int col0 = cp * 32 + l16;
        int col1 = col0 + 16;
        const float* aBase = &hTile[(rt * 16 + l16) * H_STRIDE];
        const float* bB0 = W2t + (size_t)col0 * HID;
        const float* bB1 = W2t + (size_t)col1 * HID;
        v8f acc0 = {}, acc1 = {};
        for (int k = 0; k < HID; k += 4) {
            int ka = k + 2 * lhalf;
            v2f a   = *(const v2f*)(aBase + ka);
            v2f bv0 = *(const v2f*)(bB0 + ka);
            v2f bv1 = *(const v2f*)(bB1 + ka);
            acc0 = __builtin_amdgcn_wmma_f32_16x16x4_f32(
                       false, a, false, bv0, (short)0, acc0, false, false);
            acc1 = __builtin_amdgcn_wmma_f32_16x16x4_f32(
                       false, a, false, bv1, (short)0, acc1, false, false);
        }
        if (col0 < GD) {
            float bias = b2[col0];
            for (int j = 0; j < 8; ++j) {
                int rloc = rt * 16 + j + 8 * lhalf;
                hout[(size_t)(row0 + rloc) * GD + col0] = acc0[j] + bias;
            }
        }
        if (col1 < GD) {
            float bias = b2[col1];
            for (int j = 0; j < 8; ++j) {
                int rloc = rt * 16 + j + 8 * lhalf;
                hout[(size_t)(row0 + rloc) * GD + col1] = acc1[j] + bias;
            }
        }
    }
}

// ---------------------------------------------------------------- batchnorm

__global__ __launch_bounds__(256)
void bn_stats_kernel(const float* __restrict__ h, float* __restrict__ mu,
                     float* __restrict__ s2) {
    __shared__ float smu[GD], ss2[GD];
    for (int c = threadIdx.x; c < GD; c += 256) { smu[c] = 0.0f; ss2[c] = 0.0f; }
    __syncthreads();
    int r0 = blockIdx.x * 256;
    for (int idx = threadIdx.x; idx < 256 * GD; idx += 256) {
        int r = r0 + idx / GD;
        if (r < GN) {
            int c = idx % GD;
            float v = h[(size_t)r * GD + c];
            atomicAdd(&smu[c], v);
            atomicAdd(&ss2[c], v * v);
        }
    }
    __syncthreads();
    for (int c = threadIdx.x; c < GD; c += 256) {
        unsafeAtomicAdd(&mu[c], smu[c]);
        unsafeAtomicAdd(&s2[c], ss2[c]);
    }
}

__global__ void bn_finalize_kernel(const float* mu, const float* s2,
                                   const float* gamma, const float* beta,
                                   float* scale, float* shift) {
    int c = blockIdx.x * blockDim.x + threadIdx.x;
    if (c >= GD) return;
    float m   = mu[c] * (1.0f / GN);
    float var = s2[c] * (1.0f / GN) - m * m;
    float sc  = gamma[c] * rsqrtf(var + BNEPS);
    scale[c] = sc;
    shift[c] = beta[c] - m * sc;
}

__global__ void bn_apply_kernel(float* __restrict__ h,
                                const float* __restrict__ scale,
                                const float* __restrict__ shift, int relu) {
    int tid = blockIdx.x * blockDim.x + threadIdx.x;
    if (tid >= GN * GD4) return;
    int n = tid / GD4, j = tid % GD4;
    float4 v  = ((float4*)h)[(size_t)n * GD4 + j];
    float4 sc = ((const float4*)scale)[j];
    float4 sh = ((const float4*)shift)[j];
    v.x = v.x * sc.x + sh.x; v.y = v.y * sc.y + sh.y;
    v.z = v.z * sc.z + sh.z; v.w = v.w * sc.w + sh.w;
    if (relu) {
        v.x = fmaxf(v.x, 0.0f); v.y = fmaxf(v.y, 0.0f);
        v.z = fmaxf(v.z, 0.0f); v.w = fmaxf(v.w, 0.0f);
    }
    ((float4*)h)[(size_t)n * GD4 + j] = v;
}

// ---------------------------------------------------------------- pooling

__global__ void pool_kernel(const float* __restrict__ h, const int* __restrict__ batch,
                            float* __restrict__ g) {
    int tid = blockIdx.x * blockDim.x + threadIdx.x;
    if (tid >= GN * GD4) return;
    int n = tid / GD4, j = tid % GD4;
    int b = batch[n];
    float4 v = ((const float4*)h)[(size_t)n * GD4 + j];
    float* dst = g + (size_t)b * GD + 4 * j;
    unsafeAtomicAdd(dst + 0, v.x);
    unsafeAtomicAdd(dst + 1, v.y);
    unsafeAtomicAdd(dst + 2, v.z);
    unsafeAtomicAdd(dst + 3, v.w);
}

__global__ void count_kernel(const int* __restrict__ batch, float* __restrict__ cnt) {
    int n = blockIdx.x * blockDim.x + threadIdx.x;
    if (n < GN) unsafeAtomicAdd(&cnt[batch[n]], 1.0f);
}

__global__ void pool_div_kernel(float* g, const float* cnt) {
    int i = blockIdx.x * blockDim.x + threadIdx.x;
    if (i >= GB * GD) return;
    g[i] /= fmaxf(cnt[i / GD], 1.0f);
}

// ---------------------------------------------------------------- small GEMMs

__global__ void mm_kernel(const float* __restrict__ A, const float* __restrict__ Bm,
                          const float* __restrict__ bias, float* __restrict__ C,
                          int M, int K, int Ncol, int relu) {
    int tid = blockIdx.x * blockDim.x + threadIdx.x;
    if (tid >= M * Ncol) return;
    int i = tid / Ncol, jc = tid % Ncol;
    float acc = bias ? bias[jc] : 0.0f;
    for (int k = 0; k < K; ++k)
        acc += A[(size_t)i * K + k] * Bm[(size_t)k * Ncol + jc];
    if (relu) acc = fmaxf(acc, 0.0f);
    C[(size_t)i * Ncol + jc] = acc;
}

__global__ void rownorm_kernel(const float* __restrict__ z, float* __restrict__ out) {
    int i = blockIdx.x * blockDim.x + threadIdx.x;
    if (i >= GB) return;
    float s = 0.0f;
    for (int k = 0; k < GD; ++k) { float v = z[(size_t)i * GD + k]; s += v * v; }
    float inv = 1.0f / fmaxf(sqrtf(s), 1e-12f);
    for (int k = 0; k < GD; ++k) out[(size_t)i * GD + k] = z[(size_t)i * GD + k] * inv;
}

__global__ void sym_pm_kernel(const float* __restrict__ P, float* __restrict__ pm) {
    int tid = blockIdx.x * blockDim.x + threadIdx.x;
    if (tid >= GD * GD) return;
    int i = tid / GD, j = tid % GD;
    pm[tid] = 0.5f * (P[(size_t)i * GD + j] + P[(size_t)j * GD + i]);
}

// logits[r][jc] = dot(t[r], feats[c(jc)]) / TEMP, c skipping positive column
__global__ void logits_kernel(const float* __restrict__ t,
                              const float* __restrict__ feats,
                              float* __restrict__ out) {
    int tid = blockIdx.x * blockDim.x + threadIdx.x;
    if (tid >= 1024 * 1024) return;
    int r = tid >> 10, jc = tid & 1023;
    int pos = (r + GB) & 1023;
    int c = (jc == 0) ? pos : ((jc - 1) < pos ? (jc - 1) : jc);
    const float* tr = t + (size_t)r * GD;
    const float* fc = feats + (size_t)c * GD;
    float acc = 0.0f;
    for (int k = 0; k < GD; ++k) acc += tr[k] * fc[k];
    out[tid] = acc * INV_TEMP;
}

// ---------------------------------------------------------------- launcher

extern "C" void kernel_launch(void* const* d_in, const int* in_sizes, int n_in,
                              void* d_out, int out_size, void* d_ws, size_t ws_size,
                              hipStream_t stream) {
    // params in setup_inputs() insertion order
    const float* xe1 = (const float*)d_in[0];
    const float* xe2 = (const float*)d_in[1];
    const float *ee1[5], *ee2[5], *W1[5], *b1[5], *W2[5], *b2[5], *gm[5], *bt[5];
    for (int l = 0; l < 5; ++l) {
        int base = 2 + 8 * l;
        ee1[l] = (const float*)d_in[base + 0];
        ee2[l] = (const float*)d_in[base + 1];
        W1[l]  = (const float*)d_in[base + 2];
        b1[l]  = (const float*)d_in[base + 3];
        W2[l]  = (const float*)d_in[base + 4];
        b2[l]  = (const float*)d_in[base + 5];
        gm[l]  = (const float*)d_in[base + 6];
        bt[l]  = (const float*)d_in[base + 7];
    }
    const float* pW1 = (const float*)d_in[42];
    const float* pb1 = (const float*)d_in[43];
    const float* pW2 = (const float*)d_in[44];
    const float* pb2 = (const float*)d_in[45];
    const float* P   = (const float*)d_in[46];
    const int* xv[2]    = {(const int*)d_in[47], (const int*)d_in[51]};
    const int* eidx[2]  = {(const int*)d_in[48], (const int*)d_in[52]};
    const int* eattr[2] = {(const int*)d_in[49], (const int*)d_in[53]};
    const int* batch[2] = {(const int*)d_in[50], (const int*)d_in[54]};

    float* ws    = (float*)d_ws;
    float* h     = ws;                          // 30,000,000
    float* agg   = ws + 30000000ull;            // 30,000,000
    float* mu    = ws + 60000000ull;            // 512
    float* s2    = mu + 512;                    // 512
    float* scale = s2 + 512;                    // 512
    float* shift = scale + 512;                 // 512
    float* g     = shift + 512;                 // 512*300
    float* cnt   = g + (size_t)GB * GD;         // 512
    float* z1    = cnt + 512;                   // 512*300
    float* z2    = z1 + (size_t)GB * GD;        // 512*300
    float* feats = z2 + (size_t)GB * GD;        // 1024*300
    float* pm    = feats + 2ull * GB * GD;      // 90,112 (padded)
    float* tmat  = pm + 90112ull;               // 1024*300
    float* w1t0  = tmat + 2ull * GB * GD;       // 5 x 608*300
    float* w2t0  = w1t0 + 5ull * W1PAD * GD;    // 5 x 320*600
    float* wend  = w2t0 + 5ull * W2PAD * HID;
    if (ws_size < (size_t)(wend - ws) * sizeof(float)) return;

    float* out = (float*)d_out;
    const int T = 256;
    auto blk = [](long long n) { return (int)((n + 255) / 256); };

    // pre-pack transposed/padded weights for all layers (view-invariant)
    float* w1t[5]; float* w2t[5];
    for (int l = 0; l < 5; ++l) {
        w1t[l] = w1t0 + (size_t)l * W1PAD * GD;
        w2t[l] = w2t0 + (size_t)l * W2PAD * HID;
        pack_kernel<<<blk((long long)W1PAD * GD), T, 0, stream>>>(W1[l], w1t[l], GD, HID, W1PAD);
        pack_kernel<<<blk((long long)W2PAD * HID), T, 0, stream>>>(W2[l], w2t[l], HID, GD, W2PAD);
    }

    for (int v = 0; v < 2; ++v) {
        embed_kernel<<<blk((long long)GN * GD4), T, 0, stream>>>(xv[v], xe1, xe2, h);
        for (int l = 0; l < 5; ++l) {
            init_agg_kernel<<<blk((long long)GN * GD4), T, 0, stream>>>(h, ee1[l], ee2[l], agg);
            edge_scatter_kernel<<<blk((long long)GE * GD4), T, 0, stream>>>(
                eidx[v], eattr[v], h, ee1[l], ee2[l], agg);
            gin_mlp_kernel<<<GN / TILE_ROWS, 256, 0, stream>>>(
                agg, w1t[l], b1[l], w2t[l], b2[l], h);
            zero_kernel<<<blk(1024), T, 0, stream>>>(mu, 1024);  // mu + s2 contiguous
            bn_stats_kernel<<<blk(GN), 256, 0, stream>>>(h, mu, s2);
            bn_finalize_kernel<<<2, 256, 0, stream>>>(mu, s2, gm[l], bt[l], scale, shift);
            bn_apply_kernel<<<blk((long long)GN * GD4), T, 0, stream>>>(
                h, scale, shift, (l < 4) ? 1 : 0);
        }
        zero_kernel<<<blk(GB * GD + 512), T, 0, stream>>>(g, GB * GD + 512);  // g + cnt
        pool_kernel<<<blk((long long)GN * GD4), T, 0, stream>>>(h, batch[v], g);
        count_kernel<<<blk(GN), T, 0, stream>>>(batch[v], cnt);
        pool_div_kernel<<<blk(GB * GD), T, 0, stream>>>(g, cnt);
        mm_kernel<<<blk(GB * GD), T, 0, stream>>>(g, pW1, pb1, z1, GB, GD, GD, 1);
        mm_kernel<<<blk(GB * GD), T, 0, stream>>>(z1, pW2, pb2, z2, GB, GD, GD, 0);
        rownorm_kernel<<<blk(GB), T, 0, stream>>>(z2, feats + (size_t)v * GB * GD);
    }

    sym_pm_kernel<<<blk(GD * GD), T, 0, stream>>>(P, pm);
    mm_kernel<<<blk(2 * GB * GD), T, 0, stream>>>(feats, pm, nullptr, tmat, 2 * GB, GD, GD, 0);
    logits_kernel<<<blk(1024 * 1024), T, 0, stream>>>(tmat, feats, out);
    zero_kernel<<<blk(1024), T, 0, stream>>>(out + 1024ull * 1024ull, 1024);  // labels
}